// OnlineLabelSmoothing_41927470744053
// MI455X (gfx1250) — compile-verified
//
#include <hip/hip_runtime.h>
#include <cstdint>

#define C_DIM 10000
#define B_DIM 8192
#define TPB   256

typedef float v2f __attribute__((ext_vector_type(2)));
typedef float v8f __attribute__((ext_vector_type(8)));

// ---- CDNA5 async global -> LDS copy (ASYNCcnt path, per CDNA5 ISA) ----
__device__ __forceinline__ void async_g2lds_b128(uint32_t lds_byte_off, const void* gaddr) {
    asm volatile("global_load_async_to_lds_b128 %0, %1, off"
                 :: "v"(lds_byte_off), "v"(gaddr)
                 : "memory");
}
__device__ __forceinline__ void wait_asynccnt0() {
    asm volatile("s_wait_asynccnt 0" ::: "memory");
}

// One block per row b of x. Computes per-row loss into row_loss[b], and for the
// (rare: P ~= 1/C) case pred==target, scatter-adds softmax probs into out_grad.
__global__ __launch_bounds__(TPB) void ols_row_kernel(
    const float* __restrict__ x, const int* __restrict__ target,
    const float* __restrict__ matrix, float* __restrict__ out_grad,
    float* __restrict__ out_count, float* __restrict__ row_loss)
{
    __shared__ __align__(16) float sx[C_DIM];
    __shared__ float sred[TPB];
    __shared__ int   sidx[TPB];

    const int b   = blockIdx.x;
    const int tid = threadIdx.x;
    const float* xrow = x + (size_t)b * C_DIM;

    // ---- stage x row (40 KB) into LDS with async b128 copies ----
    for (int i = tid; i < C_DIM / 4; i += TPB) {
        uint32_t loff = (uint32_t)(uintptr_t)(&sx[i * 4]);   // low 32 bits of generic addr = LDS offset
        async_g2lds_b128(loff, (const void*)(xrow + i * 4));
    }
    wait_asynccnt0();
    __syncthreads();

    // ---- pass 1: max + first-index argmax ----
    float bv = -INFINITY; int bi = 0x7fffffff;
    for (int i = tid; i < C_DIM; i += TPB) {
        float v = sx[i];
        if (v > bv || (v == bv && i < bi)) { bv = v; bi = i; }
    }
    sred[tid] = bv; sidx[tid] = bi;
    __syncthreads();
    for (int s = TPB / 2; s > 0; s >>= 1) {
        if (tid < s) {
            float ov = sred[tid + s]; int oi = sidx[tid + s];
            if (ov > sred[tid] || (ov == sred[tid] && oi < sidx[tid])) {
                sred[tid] = ov; sidx[tid] = oi;
            }
        }
        __syncthreads();
    }
    const float maxv = sred[0];
    const int   pred = sidx[0];
    __syncthreads();

    // ---- pass 2: sum(exp(x - max)) ----
    float se = 0.f;
    for (int i = tid; i < C_DIM; i += TPB) se += __expf(sx[i] - maxv);
    sred[tid] = se;
    __syncthreads();
    for (int s = TPB / 2; s > 0; s >>= 1) {
        if (tid < s) sred[tid] += sred[tid + s];
        __syncthreads();
    }
    const float logZ = maxv + __logf(sred[0]);
    __syncthreads();

    // ---- pass 3: loss_b = sum_c matrix[t][c] * (logZ - x[c])  (== -softlabel . logprob) ----
    const int t = target[b];
    const float4* mrow = (const float4*)(matrix + (size_t)t * C_DIM);
    float lacc = 0.f;
    for (int i = tid; i < C_DIM / 4; i += TPB) {
        float4 m  = mrow[i];
        float4 xv = *(const float4*)(&sx[i * 4]);
        lacc += m.x * (logZ - xv.x) + m.y * (logZ - xv.y)
              + m.z * (logZ - xv.z) + m.w * (logZ - xv.w);
    }
    sred[tid] = lacc;
    __syncthreads();
    for (int s = TPB / 2; s > 0; s >>= 1) {
        if (tid < s) sred[tid] += sred[tid + s];
        __syncthreads();
    }
    if (tid == 0) row_loss[b] = sred[0];

    // ---- rare correct-row scatter: new_grad[t] += softmax(x_b), new_count[t] += 1 ----
    if (pred == t) {
        float* grow = out_grad + (size_t)t * C_DIM;
        for (int i = tid; i < C_DIM; i += TPB)
            atomicAdd(&grow[i], __expf(sx[i] - logZ));
        if (tid == 0) atomicAdd(&out_count[t], 1.0f);
    }
}

// Deterministic mean of the 8192 per-row losses using V_WMMA_F32_16X16X4_F32.
// A = ones(16x4) => D[m,n] = sum_k B[k,n] + C[m,n]: column sums accumulate in C.
// One wave, 64 values per WMMA, 128 iterations.
__global__ __launch_bounds__(32) void loss_reduce_wmma(
    const float* __restrict__ rl, float* __restrict__ out_loss)
{
    const int lane = threadIdx.x;
    v2f a; a.x = 1.0f; a.y = 1.0f;
    v8f c = {};
    const v2f* src = (const v2f*)rl;
    for (int it = 0; it < B_DIM / 64; ++it) {
        v2f bm = src[it * 32 + lane];
        c = __builtin_amdgcn_wmma_f32_16x16x4_f32(
                /*neg_a=*/false, a, /*neg_b=*/false, bm,
                /*c_mod=*/(short)0, c, /*reuse_a=*/false, /*reuse_b=*/false);
    }
    // D VGPR0: lanes 0-15 hold row M=0, lanes 16-31 hold row M=8; both rows equal
    // the column-sum vector, so the 32-lane sum of c[0] is 2x the grand total.
    float v = c[0];
    for (int off = 16; off > 0; off >>= 1) v += __shfl_xor(v, off, 32);
    if (lane == 0) out_loss[0] = (0.5f * v) / (float)B_DIM;
}

extern "C" void kernel_launch(void* const* d_in, const int* in_sizes, int n_in,
                              void* d_out, int out_size, void* d_ws, size_t ws_size,
                              hipStream_t stream)
{
    const float* x          = (const float*)d_in[0];
    const int*   target     = (const int*)  d_in[1];
    const float* matrix     = (const float*)d_in[2];
    const float* grad_state = (const float*)d_in[3];
    const float* count      = (const float*)d_in[4];

    float* out       = (float*)d_out;              // [loss(1) | new_grad(C*C) | new_count(C)]
    float* out_loss  = out;
    float* out_grad  = out + 1;
    float* out_count = out + 1 + (size_t)C_DIM * C_DIM;
    float* row_loss  = (float*)d_ws;               // 8192 floats of scratch

    // new_grad = grad_state (+ rare atomic scatter from the kernel), new_count = count (+...)
    hipMemcpyAsync(out_grad, grad_state, sizeof(float) * (size_t)C_DIM * C_DIM,
                   hipMemcpyDeviceToDevice, stream);
    hipMemcpyAsync(out_count, count, sizeof(float) * C_DIM,
                   hipMemcpyDeviceToDevice, stream);

    ols_row_kernel<<<B_DIM, TPB, 0, stream>>>(x, target, matrix, out_grad, out_count, row_loss);
    loss_reduce_wmma<<<1, 32, 0, stream>>>(row_loss, out_loss);
}